// GCNnet_6717328851283
// MI455X (gfx1250) — compile-verified
//
#include <hip/hip_runtime.h>

// Problem constants (match reference)
#define Nn   100000
#define Ee   1600000
#define Dd   64
#define Ll   3
#define Gg   512
#define OUTC 10
#define EPSf 1e-5f

typedef __attribute__((ext_vector_type(2))) float v2f;
typedef __attribute__((ext_vector_type(8))) float v8f;

#if __has_builtin(__builtin_amdgcn_wmma_f32_16x16x4_f32)
#define HAVE_WMMA_F32X4 1
#else
#define HAVE_WMMA_F32X4 0
#endif

__device__ __forceinline__ void atomAddF(float* p, float v) {
    __hip_atomic_fetch_add(p, v, __ATOMIC_RELAXED, __HIP_MEMORY_SCOPE_AGENT);
}

// ---------------------------------------------------------------- utilities
__global__ void k_fill(float* __restrict__ p, float v, int n) {
    int i = blockIdx.x * blockDim.x + threadIdx.x;
    if (i < n) p[i] = v;
}

__global__ void k_deg_acc(const int* __restrict__ dst, float* __restrict__ deg) {
    int e = blockIdx.x * blockDim.x + threadIdx.x;
    if (e < Ee) atomAddF(&deg[dst[e]], 1.0f);
}

__global__ void k_dis(float* __restrict__ deg) {
    int i = blockIdx.x * blockDim.x + threadIdx.x;
    if (i < Nn) deg[i] = rsqrtf(deg[i]);   // deg >= 1 always (self loops)
}

__global__ void k_norm(const float* __restrict__ dis, const int* __restrict__ src,
                       const int* __restrict__ dst, float* __restrict__ nrm) {
    int i = blockIdx.x * blockDim.x + threadIdx.x;
    if (i >= Ee + Nn) return;
    int s, d;
    if (i < Ee) { s = src[i]; d = dst[i]; } else { s = d = i - Ee; }
    nrm[i] = dis[s] * dis[d];
}

// ---------------------------------------------------------------- GEMM (WMMA f32 16x16x4)
// one wave per 16-row tile; 16x64 output per wave = 4 x v8f accumulators
__global__ void __launch_bounds__(128) k_gemm(const float* __restrict__ A,
                                              const float* __restrict__ W,
                                              float* __restrict__ C, int ntiles) {
    __shared__ float sW[64 * 72];       // stride 72: bank-conflict-free B reads
    __shared__ float sA[4][16 * 68];    // stride 68: bank-conflict-free A reads

    const int tid  = threadIdx.x;
    const int w    = tid >> 5;
    const int lane = tid & 31;

    // stage W (64x64) into LDS, coalesced float4
    {
        const float4* W4 = (const float4*)W;
        for (int i = tid; i < 1024; i += 128) {
            int row = i >> 4, c4 = i & 15;
            *((float4*)&sW[row * 72 + c4 * 4]) = W4[i];
        }
    }

    const int  tile  = blockIdx.x * 4 + w;
    const bool valid = tile < ntiles;
    const int  m0    = tile * 16;
    if (valid) {
        // stage this wave's 16x64 A tile, coalesced float4 (8 per lane)
        for (int it = 0; it < 8; ++it) {
            int j = lane + 32 * it;
            int row = j >> 4, c4 = j & 15;
            float4 v = ((const float4*)(A + (size_t)(m0 + row) * 64))[c4];
            *((float4*)&sA[w][row * 68 + c4 * 4]) = v;
        }
    }
    __syncthreads();
    if (!valid) return;

    const int half = lane >> 4;       // ISA lane-half for 16x16 fragments
    const int r    = lane & 15;
    const float* Arow = &sA[w][r * 68];

#if HAVE_WMMA_F32X4
    v8f acc0 = {}, acc1 = {}, acc2 = {}, acc3 = {};
    for (int kb = 0; kb < 16; ++kb) {
        const int k0 = kb * 4 + 2 * half;           // A: V0=K0/K2, V1=K1/K3 per half
        v2f a;
        a.x = Arow[k0];
        a.y = Arow[k0 + 1];
        const float* B0 = &sW[(size_t)k0 * 72 + r];
        const float* B1 = &sW[(size_t)(k0 + 1) * 72 + r];
        v2f b;
        b.x = B0[0];   b.y = B1[0];
        acc0 = __builtin_amdgcn_wmma_f32_16x16x4_f32(false, a, false, b, (short)0, acc0, false, false);
        b.x = B0[16];  b.y = B1[16];
        acc1 = __builtin_amdgcn_wmma_f32_16x16x4_f32(false, a, false, b, (short)0, acc1, false, false);
        b.x = B0[32];  b.y = B1[32];
        acc2 = __builtin_amdgcn_wmma_f32_16x16x4_f32(false, a, false, b, (short)0, acc2, false, false);
        b.x = B0[48];  b.y = B1[48];
        acc3 = __builtin_amdgcn_wmma_f32_16x16x4_f32(false, a, false, b, (short)0, acc3, false, false);
    }
    union U { v8f v; float f[8]; };
    U u0, u1, u2, u3;
    u0.v = acc0; u1.v = acc1; u2.v = acc2; u3.v = acc3;
    for (int v = 0; v < 8; ++v) {                    // C: VGPR v -> M=v (+8 per half), N=r
        float* row = C + (size_t)(m0 + v + 8 * half) * 64 + r;
        row[0]  = u0.f[v];
        row[16] = u1.f[v];
        row[32] = u2.f[v];
        row[48] = u3.f[v];
    }
#else
    float acc[4][8] = {};
    for (int k = 0; k < 64; ++k) {
        float w0 = sW[(size_t)k * 72 + r];
        float w1 = sW[(size_t)k * 72 + 16 + r];
        float w2 = sW[(size_t)k * 72 + 32 + r];
        float w3 = sW[(size_t)k * 72 + 48 + r];
        for (int v = 0; v < 8; ++v) {
            float av = sA[w][(v + 8 * half) * 68 + k];
            acc[0][v] += av * w0; acc[1][v] += av * w1;
            acc[2][v] += av * w2; acc[3][v] += av * w3;
        }
    }
    for (int v = 0; v < 8; ++v) {
        float* row = C + (size_t)(m0 + v + 8 * half) * 64 + r;
        row[0] = acc[0][v]; row[16] = acc[1][v]; row[32] = acc[2][v]; row[48] = acc[3][v];
    }
#endif
}

// ---------------------------------------------------------------- message scatter
__global__ void k_agg_init(float* __restrict__ agg, const float* __restrict__ b) {
    int i = blockIdx.x * blockDim.x + threadIdx.x;  // float4 index
    if (i >= Nn * 16) return;
    int c0 = (i & 15) * 4;
    ((float4*)agg)[i] = make_float4(b[c0], b[c0 + 1], b[c0 + 2], b[c0 + 3]);
}

__global__ void k_scatter(const float* __restrict__ h, const float* __restrict__ nrm,
                          const int* __restrict__ src, const int* __restrict__ dst,
                          float* __restrict__ agg) {
    long long idx = (long long)blockIdx.x * blockDim.x + threadIdx.x;
    int edge = (int)(idx >> 5);
    int lane = (int)(idx & 31);
    if (edge >= Ee + Nn) return;
    int s, d;
    if (edge < Ee) { s = src[edge]; d = dst[edge]; } else { s = d = edge - Ee; }
    float nm = nrm[edge];
    float2 hv = ((const float2*)h)[(size_t)s * 32 + lane];
    float* ap = agg + (size_t)d * 64 + lane * 2;
    atomAddF(ap,     hv.x * nm);
    atomAddF(ap + 1, hv.y * nm);
}

// ---------------------------------------------------------------- BatchNorm
__global__ void k_bn_stats(const float* __restrict__ agg, float* __restrict__ stats) {
    __shared__ float ls[256], lss[256];
    int tid = threadIdx.x;
    int c = tid & 63, g = tid >> 6;
    float s = 0.f, sq = 0.f;
    for (int r = blockIdx.x * 4 + g; r < Nn; r += gridDim.x * 4) {
        float v = agg[(size_t)r * 64 + c];
        s += v; sq += v * v;
    }
    ls[tid] = s; lss[tid] = sq;
    __syncthreads();
    if (tid < 64) {
        s  = ls[tid]  + ls[tid + 64]  + ls[tid + 128]  + ls[tid + 192];
        sq = lss[tid] + lss[tid + 64] + lss[tid + 128] + lss[tid + 192];
        atomAddF(&stats[tid], s);
        atomAddF(&stats[64 + tid], sq);
    }
}

__global__ void k_bn_final(float* __restrict__ stats, const float* __restrict__ gamma,
                           const float* __restrict__ beta) {
    int c = threadIdx.x;
    if (c >= 64) return;
    float mean = stats[c] * (1.0f / Nn);
    float var  = stats[64 + c] * (1.0f / Nn) - mean * mean;
    float sc   = gamma[c] * rsqrtf(var + EPSf);
    stats[128 + c] = sc;
    stats[192 + c] = beta[c] - mean * sc;
}

__global__ void k_bn_apply(float* __restrict__ agg, const float* __restrict__ stats) {
    int i = blockIdx.x * blockDim.x + threadIdx.x;  // float4 index
    if (i >= Nn * 16) return;
    int c0 = (i & 15) * 4;
    const float* sc = stats + 128;
    const float* sh = stats + 192;
    float4 v = ((float4*)agg)[i];
    v.x = fmaxf(v.x * sc[c0]     + sh[c0],     0.f);
    v.y = fmaxf(v.y * sc[c0 + 1] + sh[c0 + 1], 0.f);
    v.z = fmaxf(v.z * sc[c0 + 2] + sh[c0 + 2], 0.f);
    v.w = fmaxf(v.w * sc[c0 + 3] + sh[c0 + 3], 0.f);
    ((float4*)agg)[i] = v;
}

// ---------------------------------------------------------------- pooling + head
__global__ void k_pool(const float* __restrict__ h, const int* __restrict__ batch,
                       float* __restrict__ pooled, float* __restrict__ cnt) {
    long long idx = (long long)blockIdx.x * blockDim.x + threadIdx.x;
    int node = (int)(idx >> 5);
    int lane = (int)(idx & 31);
    if (node >= Nn) return;
    int b = batch[node];
    float2 hv = ((const float2*)h)[(size_t)node * 32 + lane];
    float* pp = pooled + (size_t)b * 64 + lane * 2;
    atomAddF(pp,     hv.x);
    atomAddF(pp + 1, hv.y);
    if (lane == 0) atomAddF(&cnt[b], 1.f);
}

__global__ void k_head(const float* __restrict__ pooled, const float* __restrict__ cnt,
                       const float* __restrict__ Wf, const float* __restrict__ bf,
                       float* __restrict__ out) {
    int i = blockIdx.x * blockDim.x + threadIdx.x;
    if (i >= Gg * OUTC) return;
    int g = i / OUTC, o = i - g * OUTC;
    float inv = 1.0f / fmaxf(cnt[g], 1.0f);
    const float* pg = pooled + (size_t)g * 64;
    float acc = 0.f;
    for (int k = 0; k < 64; ++k) acc += pg[k] * Wf[k * OUTC + o];
    out[i] = bf[o] + acc * inv;
}

// ---------------------------------------------------------------- driver
extern "C" void kernel_launch(void* const* d_in, const int* in_sizes, int n_in,
                              void* d_out, int out_size, void* d_ws, size_t ws_size,
                              hipStream_t stream) {
    const float* x     = (const float*)d_in[0];
    const int*   ei    = (const int*)  d_in[1];
    const int*   batch = (const int*)  d_in[2];
    const float* Ws    = (const float*)d_in[3];
    const float* bs    = (const float*)d_in[4];
    const float* gam   = (const float*)d_in[5];
    const float* bet   = (const float*)d_in[6];
    const float* Wf    = (const float*)d_in[7];
    const float* bf    = (const float*)d_in[8];
    float*       out   = (float*)d_out;

    const int* e_src = ei;
    const int* e_dst = ei + Ee;

    float* ws     = (float*)d_ws;
    float* dis    = ws;                          // N (holds deg, then rsqrt(deg))
    float* nrm    = dis + Nn;                    // E+N
    float* hW     = nrm + (Ee + Nn);             // N*D
    float* agg    = hW + (size_t)Nn * Dd;        // N*D (also next-layer input)
    float* stats  = agg + (size_t)Nn * Dd;       // 256: sum|sumsq|scale|shift
    float* pooled = stats + 256;                 // G*D
    float* cnt    = pooled + (size_t)Gg * Dd;    // G (contiguous after pooled)

    const int B = 256;

    // degree (self-loop = init 1), rsqrt, edge norms
    k_fill   <<<(Nn + B - 1) / B, B, 0, stream>>>(dis, 1.0f, Nn);
    k_deg_acc<<<(Ee + B - 1) / B, B, 0, stream>>>(e_dst, dis);
    k_dis    <<<(Nn + B - 1) / B, B, 0, stream>>>(dis);
    k_norm   <<<(Ee + Nn + B - 1) / B, B, 0, stream>>>(dis, e_src, e_dst, nrm);

    const int ntiles = Nn / 16;                  // 6250
    const float* hIn = x;
    for (int l = 0; l < Ll; ++l) {
        k_gemm    <<<(ntiles + 3) / 4, 128, 0, stream>>>(hIn, Ws + (size_t)l * Dd * Dd, hW, ntiles);
        k_agg_init<<<(Nn * 16 + B - 1) / B, B, 0, stream>>>(agg, bs + l * Dd);
        long long sthreads = (long long)(Ee + Nn) * 32;
        k_scatter <<<(unsigned)((sthreads + B - 1) / B), B, 0, stream>>>(hW, nrm, e_src, e_dst, agg);
        k_fill    <<<1, 256, 0, stream>>>(stats, 0.f, 128);
        k_bn_stats<<<250, 256, 0, stream>>>(agg, stats);
        k_bn_final<<<1, 64, 0, stream>>>(stats, gam + l * Dd, bet + l * Dd);
        k_bn_apply<<<(Nn * 16 + B - 1) / B, B, 0, stream>>>(agg, stats);
        hIn = agg;                               // in-place ping: stream order makes this safe
    }

    k_fill<<<(Gg * Dd + Gg + B - 1) / B, B, 0, stream>>>(pooled, 0.f, Gg * Dd + Gg);
    long long pthreads = (long long)Nn * 32;
    k_pool<<<(unsigned)((pthreads + B - 1) / B), B, 0, stream>>>(agg, batch, pooled, cnt);
    k_head<<<(Gg * OUTC + B - 1) / B, B, 0, stream>>>(pooled, cnt, Wf, bf, out);
}